// GeoGCN_56581899157894
// MI455X (gfx1250) — compile-verified
//
#include <hip/hip_runtime.h>

#define IN_F   16
#define HID_F  128
#define OUT_F  12
#define BN_EPS 1e-5f

typedef __attribute__((ext_vector_type(2))) float v2f;
typedef __attribute__((ext_vector_type(8))) float v8f;

// CDNA5 full-precision WMMA: D(16x16,f32) = A(16x4,f32) * B(4x16,f32) + C
__device__ __forceinline__ v8f wmma4(v2f a, v2f b, v8f c) {
    return __builtin_amdgcn_wmma_f32_16x16x4_f32(
        /*neg_a=*/false, a, /*neg_b=*/false, b,
        /*c_mod=*/(short)0, c, /*reuse_a=*/false, /*reuse_b=*/false);
}

// native fp32 atomic add (global_atomic_add_f32, no CAS loop)
__device__ __forceinline__ void atomAddF(float* p, float v) {
    unsafeAtomicAdd(p, v);
}

// ---------------- degree / normalization ----------------

__global__ void k_deg_init(float* deg, int n) {
    int i = blockIdx.x * blockDim.x + threadIdx.x;
    if (i < n) deg[i] = 1.0f;   // self-loop weight
}

__global__ void k_deg_scatter(const long long* __restrict__ ei,
                              const float* __restrict__ w, float* deg, int e) {
    int i = blockIdx.x * blockDim.x + threadIdx.x;
    if (i < e) atomAddF(&deg[(int)ei[(size_t)e + i]], w[i]);   // dst row
}

__global__ void k_dinv(const float* __restrict__ deg, float* dinv, int n) {
    int i = blockIdx.x * blockDim.x + threadIdx.x;
    if (i < n) { float d = deg[i]; dinv[i] = d > 0.0f ? rsqrtf(d) : 0.0f; }
}

__global__ void k_norm(const long long* __restrict__ ei,
                       const float* __restrict__ w,
                       const float* __restrict__ dinv, float* normv, int e) {
    int i = blockIdx.x * blockDim.x + threadIdx.x;
    if (i < e) {
        int s = (int)ei[i];
        int d = (int)ei[(size_t)e + i];
        normv[i] = dinv[s] * w[i] * dinv[d];
    }
}

// ---------------- WMMA GEMMs (fp32) ----------------

// H[N,128] = X[N,16] @ W0[16,128]; fused: AGG = H * dinv^2 (self-loop init)
__global__ __launch_bounds__(32)
void k_gemm_xw0(const float* __restrict__ X, const float* __restrict__ W,
                const float* __restrict__ dinv,
                float* __restrict__ H, float* __restrict__ AGG, int n) {
    int lane = threadIdx.x;
    int half = lane >> 4;       // 0: lanes 0-15, 1: lanes 16-31
    int l16  = lane & 15;
    int m0 = blockIdx.x * 16;
    int n0 = blockIdx.y * 16;
    int row  = m0 + l16;
    int rowc = row < n ? row : n - 1;
    int col  = n0 + l16;
    v8f acc = {};
    #pragma unroll
    for (int k0 = 0; k0 < IN_F; k0 += 4) {
        int ka = k0 + half * 2;
        v2f a, b;
        a.x = X[(size_t)rowc * IN_F + ka];
        a.y = X[(size_t)rowc * IN_F + ka + 1];
        b.x = W[(size_t)ka * HID_F + col];
        b.y = W[(size_t)(ka + 1) * HID_F + col];
        acc = wmma4(a, b, acc);
    }
    if (m0 + 16 <= n) {     // fast path: whole tile in range, straight stores
        #pragma unroll
        for (int r = 0; r < 8; ++r) {
            int orow = m0 + r + half * 8;
            float di = dinv[orow];
            size_t o = (size_t)orow * HID_F + col;
            H[o]   = acc[r];
            AGG[o] = acc[r] * di * di;
        }
    } else {
        #pragma unroll
        for (int r = 0; r < 8; ++r) {
            int orow = m0 + r + half * 8;
            if (orow < n) {
                float di = dinv[orow];
                size_t o = (size_t)orow * HID_F + col;
                H[o]   = acc[r];
                AGG[o] = acc[r] * di * di;
            }
        }
    }
}

// O[N,128] = H[N,128] @ W1[128,128]   4 waves per block (4 N-tiles)
__global__ __launch_bounds__(128)
void k_gemm_hw1(const float* __restrict__ H, const float* __restrict__ W,
                float* __restrict__ O, int n) {
    int lane = threadIdx.x;
    int half = lane >> 4;
    int l16  = lane & 15;
    int m0 = blockIdx.x * 16;
    int n0 = (blockIdx.y * 4 + threadIdx.y) * 16;
    int row  = m0 + l16;
    int rowc = row < n ? row : n - 1;
    int col  = n0 + l16;
    v8f acc = {};
    #pragma unroll 8
    for (int k0 = 0; k0 < HID_F; k0 += 4) {
        int ka = k0 + half * 2;
        v2f a, b;
        a.x = H[(size_t)rowc * HID_F + ka];
        a.y = H[(size_t)rowc * HID_F + ka + 1];
        b.x = W[(size_t)ka * HID_F + col];
        b.y = W[(size_t)(ka + 1) * HID_F + col];
        acc = wmma4(a, b, acc);
    }
    if (m0 + 16 <= n) {
        #pragma unroll
        for (int r = 0; r < 8; ++r)
            O[(size_t)(m0 + r + half * 8) * HID_F + col] = acc[r];
    } else {
        #pragma unroll
        for (int r = 0; r < 8; ++r) {
            int orow = m0 + r + half * 8;
            if (orow < n) O[(size_t)orow * HID_F + col] = acc[r];
        }
    }
}

// H2[N,12] = H[N,128] @ W2[128,12]; fused: out = b2 + H2 * dinv^2
__global__ __launch_bounds__(32)
void k_gemm_hw2(const float* __restrict__ H, const float* __restrict__ W,
                const float* __restrict__ dinv, const float* __restrict__ b2,
                float* __restrict__ H2, float* __restrict__ out, int n) {
    int lane = threadIdx.x;
    int half = lane >> 4;
    int l16  = lane & 15;
    int m0 = blockIdx.x * 16;
    int row  = m0 + l16;
    int rowc = row < n ? row : n - 1;
    int col  = l16;
    int ccl  = col < OUT_F ? col : OUT_F - 1;
    float cm = col < OUT_F ? 1.0f : 0.0f;
    v8f acc = {};
    #pragma unroll 8
    for (int k0 = 0; k0 < HID_F; k0 += 4) {
        int ka = k0 + half * 2;
        v2f a, b;
        a.x = H[(size_t)rowc * HID_F + ka];
        a.y = H[(size_t)rowc * HID_F + ka + 1];
        b.x = W[(size_t)ka * OUT_F + ccl] * cm;
        b.y = W[(size_t)(ka + 1) * OUT_F + ccl] * cm;
        acc = wmma4(a, b, acc);
    }
    float b2v = b2[ccl];
    #pragma unroll
    for (int r = 0; r < 8; ++r) {
        int orow = m0 + r + half * 8;
        if (orow < n && col < OUT_F) {
            float di = dinv[orow];
            size_t o = (size_t)orow * OUT_F + col;
            H2[o]  = acc[r];
            out[o] = b2v + acc[r] * di * di;
        }
    }
}

// ---------------- aggregation (SpMM scatter) ----------------

// AGG[i,c] = H[i,c] * dinv[i]^2  (layer-1 init; can't fuse: in-place hazard)
__global__ void k_agg_init128(const float* __restrict__ H,
                              const float* __restrict__ dinv,
                              float* __restrict__ AGG, int n) {
    int idx = blockIdx.x * blockDim.x + threadIdx.x;
    if (idx < n * HID_F) {
        float di = dinv[idx >> 7];
        AGG[idx] = H[idx] * di * di;
    }
}

// one block per edge; 128 lanes cover the feature dim (coalesced, L2-resident)
__global__ __launch_bounds__(128)
void k_scatter128(const float* __restrict__ H, const long long* __restrict__ ei,
                  const float* __restrict__ normv, float* __restrict__ AGG, int e) {
    int eidx = blockIdx.x;
    int c = threadIdx.x;
    int s = (int)ei[eidx];
    int d = (int)ei[(size_t)e + eidx];
    float w = normv[eidx];
    atomAddF(&AGG[(size_t)d * HID_F + c], H[(size_t)s * HID_F + c] * w);
}

// ---------------- BatchNorm + ReLU ----------------

__global__ void k_zero_stats(float* stats) {
    stats[threadIdx.x] = 0.0f;        // 256 = 128 sums + 128 sumsq
}

__global__ __launch_bounds__(128)
void k_bn_stats(const float* __restrict__ H, float* __restrict__ stats, int n) {
    int c  = threadIdx.x;             // column
    int r0 = blockIdx.x * 256;
    int r1 = r0 + 256 < n ? r0 + 256 : n;
    float s = 0.0f, q = 0.0f;
    for (int r = r0; r < r1; ++r) {
        float v = H[(size_t)r * HID_F + c];
        s += v; q += v * v;
    }
    atomAddF(&stats[c], s);
    atomAddF(&stats[HID_F + c], q);
}

__global__ void k_bn_apply(const float* __restrict__ H,
                           const float* __restrict__ stats,
                           const float* __restrict__ g,
                           const float* __restrict__ be,
                           float* __restrict__ O, int n) {
    int idx = blockIdx.x * blockDim.x + threadIdx.x;
    if (idx < n * HID_F) {
        int c = idx & (HID_F - 1);
        float invN = 1.0f / (float)n;
        float m  = stats[c] * invN;
        float va = stats[HID_F + c] * invN - m * m;
        float r  = (H[idx] - m) * rsqrtf(va + BN_EPS) * g[c] + be[c];
        O[idx] = r > 0.0f ? r : 0.0f;
    }
}

// ---------------- final 12-wide scatter ----------------

__global__ void k_scatter12(const float* __restrict__ H2,
                            const long long* __restrict__ ei,
                            const float* __restrict__ normv,
                            float* __restrict__ out, int e) {
    int i = blockIdx.x * blockDim.x + threadIdx.x;
    if (i < e) {
        int s = (int)ei[i];
        int d = (int)ei[(size_t)e + i];
        float w = normv[i];
        #pragma unroll
        for (int c = 0; c < OUT_F; ++c)
            atomAddF(&out[(size_t)d * OUT_F + c], H2[(size_t)s * OUT_F + c] * w);
    }
}

// ---------------- host launcher ----------------

static inline size_t al256(size_t x) { return (x + 255) & ~(size_t)255; }

extern "C" void kernel_launch(void* const* d_in, const int* in_sizes, int n_in,
                              void* d_out, int out_size, void* d_ws, size_t ws_size,
                              hipStream_t stream) {
    (void)n_in; (void)out_size; (void)ws_size;

    const float*     x   = (const float*)d_in[0];
    const long long* ei  = (const long long*)d_in[1];   // [2,E] int64
    const float*     ew  = (const float*)d_in[2];
    const float*     W0  = (const float*)d_in[3];
    // b0 (d_in[4]) and b1 (d_in[6]) cancel through BatchNorm mean subtraction
    const float*     W1  = (const float*)d_in[5];
    const float*     W2  = (const float*)d_in[7];
    const float*     b2  = (const float*)d_in[8];
    const float*     g0  = (const float*)d_in[9];
    const float*     be0 = (const float*)d_in[10];
    const float*     g1  = (const float*)d_in[11];
    const float*     be1 = (const float*)d_in[12];

    const int N = in_sizes[0] / IN_F;
    const int E = in_sizes[2];

    // workspace carve-up (2 x 51.2MB ping-pong + edge norm + per-node scalars)
    char* w = (char*)d_ws;
    const size_t nbBig = al256((size_t)N * HID_F * sizeof(float));
    const size_t nbE   = al256((size_t)E * sizeof(float));
    const size_t nbN   = al256((size_t)N * sizeof(float));
    float* bufA  = (float*)(w);
    float* bufB  = (float*)(w + nbBig);
    float* normv = (float*)(w + 2 * nbBig);
    float* deg   = (float*)(w + 2 * nbBig + nbE);
    float* dinv  = (float*)(w + 2 * nbBig + nbE + nbN);
    float* stats = (float*)(w + 2 * nbBig + nbE + 2 * nbN);
    float* out   = (float*)d_out;

    const int mt    = (N + 15) / 16;              // 16-row tiles
    const int gN    = (N + 255) / 256;
    const int gE    = (E + 255) / 256;
    const int gNF   = (N * HID_F + 255) / 256;
    const int gStat = (N + 255) / 256;

    // ---- normalization ----
    k_deg_init   <<<gN, 256, 0, stream>>>(deg, N);
    k_deg_scatter<<<gE, 256, 0, stream>>>(ei, ew, deg, E);
    k_dinv       <<<gN, 256, 0, stream>>>(deg, dinv, N);
    k_norm       <<<gE, 256, 0, stream>>>(ei, ew, dinv, normv, E);

    // ---- layer 0: X@W0 (+fused AGG init) -> scatter -> BN+ReLU ----
    k_gemm_xw0   <<<dim3(mt, HID_F / 16), 32, 0, stream>>>(x, W0, dinv, bufA, bufB, N);
    k_scatter128 <<<E, 128, 0, stream>>>(bufA, ei, normv, bufB, E);
    k_zero_stats <<<1, 256, 0, stream>>>(stats);
    k_bn_stats   <<<gStat, 128, 0, stream>>>(bufB, stats, N);
    k_bn_apply   <<<gNF, 256, 0, stream>>>(bufB, stats, g0, be0, bufA, N);

    // ---- layer 1: H@W1 -> AGG init -> scatter -> BN+ReLU ----
    k_gemm_hw1   <<<dim3(mt, 2), dim3(32, 4), 0, stream>>>(bufA, W1, bufB, N);
    k_agg_init128<<<gNF, 256, 0, stream>>>(bufB, dinv, bufA, N);
    k_scatter128 <<<E, 128, 0, stream>>>(bufB, ei, normv, bufA, E);
    k_zero_stats <<<1, 256, 0, stream>>>(stats);
    k_bn_stats   <<<gStat, 128, 0, stream>>>(bufA, stats, N);
    k_bn_apply   <<<gNF, 256, 0, stream>>>(bufA, stats, g1, be1, bufB, N);

    // ---- layer 2: H@W2 (+fused out init with b2) -> scatter into d_out ----
    k_gemm_hw2   <<<mt, 32, 0, stream>>>(bufB, W2, dinv, b2, bufA, out, N);
    k_scatter12  <<<gE, 256, 0, stream>>>(bufA, ei, normv, out, E);
}